// INTP_Model_59871844107120
// MI455X (gfx1250) — compile-verified
//
#include <hip/hip_runtime.h>
#include <math.h>

// ---------------------------------------------------------------------------
// CDNA5 (gfx1250) fp32 WMMA implementation of the INTP transformer model.
// All matmuls use V_WMMA_F32_16X16X4_F32 (wave32, 16x16 tile, K-step 4).
// The rp-MLP on `dams` is folded algebraically: a = d0*W0 + d1*W1 + c with
// W = rp_w1@rp_w2, so attention needs only three Gram matrices S_w = K@(Q.w)^T
// instead of materializing the (b,n,n,128) tensor.
// ---------------------------------------------------------------------------

typedef float v2f __attribute__((ext_vector_type(2)));
typedef float v8f __attribute__((ext_vector_type(8)));

#define LN_EPS 1e-6f

__device__ __forceinline__ v8f wmma4(v2f a, v2f b, v8f c) {
  // (neg_a, A, neg_b, B, c_mod, C, reuse_a, reuse_b)
  return __builtin_amdgcn_wmma_f32_16x16x4_f32(false, a, false, b, (short)0, c,
                                               false, false);
}

// ---------------- generic GEMM: C = act(A@B + bias) -------------------------
// A: MxK row-major, B: KxN row-major, bias: length-N or nullptr.
// grid: (ceil(N/128), M/16)  block: 256 threads = 8 waves, 1 tile per wave.
// ACT: 0=none, 1=relu, 2=exact gelu
template <int ACT>
__global__ void gemm_kernel(const float* __restrict__ A,
                            const float* __restrict__ B,
                            const float* __restrict__ bias,
                            float* __restrict__ C, int M, int N, int K) {
  const int wave = threadIdx.x >> 5;
  const int lane = threadIdx.x & 31;
  const int half = lane >> 4;       // 0 or 1 -> k-pair selector
  const int l16  = lane & 15;
  const int tn = blockIdx.x * 8 + wave;
  const int tm = blockIdx.y;
  const int row_a = tm * 16 + l16;  // A fragment row (M is always mult of 16)
  const int col_b = tn * 16 + l16;  // B fragment / store column
  const bool cb_ok = (col_b < N);

  v8f acc = {0.f, 0.f, 0.f, 0.f, 0.f, 0.f, 0.f, 0.f};
  for (int k = 0; k < K; k += 4) {
    const int ka = k + half * 2;
    v2f a, b;
    a.x = (ka     < K) ? A[(size_t)row_a * K + ka]     : 0.f;
    a.y = (ka + 1 < K) ? A[(size_t)row_a * K + ka + 1] : 0.f;
    b.x = (ka     < K && cb_ok) ? B[(size_t)ka * N + col_b]       : 0.f;
    b.y = (ka + 1 < K && cb_ok) ? B[(size_t)(ka + 1) * N + col_b] : 0.f;
    acc = wmma4(a, b, acc);
  }

  const float bv = (bias != nullptr && cb_ok) ? bias[col_b] : 0.f;
#pragma unroll
  for (int i = 0; i < 8; ++i) {
    float v = acc[i] + bv;
    if (ACT == 1) v = fmaxf(v, 0.f);
    if (ACT == 2) v = 0.5f * v * (1.f + erff(v * 0.70710678118654752f));
    const int rowc = tm * 16 + i + half * 8;  // D layout: VGPR i -> M=i / i+8
    if (cb_ok) C[(size_t)rowc * N + col_b] = v;
  }
}

// ---------------- LayerNorm: out = LN(prescale*in (+res)) * g + b -----------
// grid: rows, block: 128 (d_model)
__global__ void ln_kernel(const float* __restrict__ in,
                          const float* __restrict__ res, float prescale,
                          const float* __restrict__ g,
                          const float* __restrict__ b,
                          float* __restrict__ out) {
  const int row = blockIdx.x;
  const int t = threadIdx.x;
  __shared__ float sd[128];
  float v = in[(size_t)row * 128 + t] * prescale;
  if (res != nullptr) v += res[(size_t)row * 128 + t];
  sd[t] = v;
  __syncthreads();
  for (int s = 64; s > 0; s >>= 1) {
    if (t < s) sd[t] += sd[t + s];
    __syncthreads();
  }
  const float mu = sd[0] * (1.f / 128.f);
  __syncthreads();
  const float dv = v - mu;
  sd[t] = dv * dv;
  __syncthreads();
  for (int s = 64; s > 0; s >>= 1) {
    if (t < s) sd[t] += sd[t + s];
    __syncthreads();
  }
  const float var = sd[0] * (1.f / 128.f);
  out[(size_t)row * 128 + t] = dv / sqrtf(var + LN_EPS) * g[t] + b[t];
}

// ---------------- rp fold: W(2x128)=rp_w1@rp_w2, C=rp_b1@rp_w2+rp_b2 --------
__global__ void rpfold_kernel(const float* __restrict__ w1,
                              const float* __restrict__ b1,
                              const float* __restrict__ w2,
                              const float* __restrict__ b2,
                              float* __restrict__ W, float* __restrict__ C) {
  const int dd = threadIdx.x;  // 128 threads
  float a0 = 0.f, a1 = 0.f, cc = 0.f;
  for (int j = 0; j < 128; ++j) {
    const float r = w2[j * 128 + dd];
    a0 += w1[j] * r;
    a1 += w1[128 + j] * r;
    cc += b1[j] * r;
  }
  W[dd] = a0;
  W[128 + dd] = a1;
  C[dd] = cc + b2[dd];
}

// ---------------- fused attention -------------------------------------------
// grid: (n/16 k-tiles, 4 heads, 2 batch), block 256 (8 waves).
// Stage 1: each wave -> 2 q-tiles, 3 WMMA accumulators (S0,S1,Sc over d=128),
//          combine with dams + mask -> LDS score tile [16 k x 256 q].
// Stage 2: softmax over q (axis matching reference softmax on 'bhkq' last dim).
// Stage 3: O = attn @ V, each wave one 16-wide d tile, WMMA over q=256.
__global__ void attn_kernel(const float* __restrict__ Q,
                            const float* __restrict__ Km,
                            const float* __restrict__ V,
                            const float* __restrict__ dams,
                            const int* __restrict__ amask,
                            const float* __restrict__ W2,
                            const float* __restrict__ Cc,
                            float* __restrict__ O) {
  const int n = 256, HD = 512;
  const int bb = blockIdx.z, h = blockIdx.y, kt = blockIdx.x;
  const int wave = threadIdx.x >> 5, lane = threadIdx.x & 31;
  const int half = lane >> 4, l16 = lane & 15;

  __shared__ float s_att[16][272];  // 16 k-rows x 256 q (+pad), ~17 KB
  __shared__ float s_red[16][16];

  const float* Qh = Q + (size_t)bb * n * HD + h * 128;
  const float* Kh = Km + (size_t)bb * n * HD + h * 128;
  const float* Vh = V + (size_t)bb * n * HD + h * 128;

  const int k_row = kt * 16 + l16;

  for (int t = 0; t < 2; ++t) {
    const int qt = wave * 2 + t;
    const int q_col = qt * 16 + l16;
    v8f acc0 = {0.f, 0.f, 0.f, 0.f, 0.f, 0.f, 0.f, 0.f};
    v8f acc1 = {0.f, 0.f, 0.f, 0.f, 0.f, 0.f, 0.f, 0.f};
    v8f accc = {0.f, 0.f, 0.f, 0.f, 0.f, 0.f, 0.f, 0.f};
    for (int dd = 0; dd < 128; dd += 4) {
      const int ka = dd + half * 2;
      const v2f a  = *(const v2f*)(Kh + (size_t)k_row * HD + ka);
      const v2f qv = *(const v2f*)(Qh + (size_t)q_col * HD + ka);
      const v2f w0 = *(const v2f*)(W2 + ka);
      const v2f w1 = *(const v2f*)(W2 + 128 + ka);
      const v2f cw = *(const v2f*)(Cc + ka);
      acc0 = wmma4(a, qv * w0, acc0);
      acc1 = wmma4(a, qv * w1, acc1);
      accc = wmma4(a, qv * cw, accc);
    }
    const float invT = 0.088388347648318447f;  // 1/sqrt(128)
#pragma unroll
    for (int i = 0; i < 8; ++i) {
      const int kloc = i + half * 8;
      const int krow = kt * 16 + kloc;
      const size_t base = (size_t)(bb * n + krow) * n + q_col;
      const float d0 = dams[base * 2 + 0];
      const float d1 = dams[base * 2 + 1];
      float vv = (d0 * acc0[i] + d1 * acc1[i] + accc[i]) * invT;
      if (amask[base] == 0) vv = -1e10f;
      s_att[kloc][q_col] = vv;
    }
  }
  __syncthreads();

  // softmax over q (256) per k-row; 16 threads per row
  const int r = threadIdx.x >> 4;
  const int j = threadIdx.x & 15;
  float m = -3.0e38f;
  for (int qq = j; qq < n; qq += 16) m = fmaxf(m, s_att[r][qq]);
  s_red[r][j] = m;
  __syncthreads();
  if (j == 0) {
    float mm = s_red[r][0];
    for (int t2 = 1; t2 < 16; ++t2) mm = fmaxf(mm, s_red[r][t2]);
    s_red[r][0] = mm;
  }
  __syncthreads();
  const float rowmax = s_red[r][0];
  __syncthreads();
  float ssum = 0.f;
  for (int qq = j; qq < n; qq += 16) {
    const float e = expf(s_att[r][qq] - rowmax);
    s_att[r][qq] = e;
    ssum += e;
  }
  s_red[r][j] = ssum;
  __syncthreads();
  if (j == 0) {
    float st = 0.f;
    for (int t2 = 0; t2 < 16; ++t2) st += s_red[r][t2];
    s_red[r][0] = st;
  }
  __syncthreads();
  const float inv = 1.f / s_red[r][0];
  for (int qq = j; qq < n; qq += 16) s_att[r][qq] *= inv;
  __syncthreads();

  // O[k, dcol] = sum_q attn[k,q] * V[q,dcol]; wave -> one 16-wide d tile
  const int dcol = wave * 16 + l16;
  v8f acc = {0.f, 0.f, 0.f, 0.f, 0.f, 0.f, 0.f, 0.f};
  for (int qq = 0; qq < n; qq += 4) {
    const int ka = qq + half * 2;
    v2f a, b;
    a.x = s_att[l16][ka];
    a.y = s_att[l16][ka + 1];
    b.x = Vh[(size_t)ka * HD + dcol];
    b.y = Vh[(size_t)(ka + 1) * HD + dcol];
    acc = wmma4(a, b, acc);
  }
#pragma unroll
  for (int i = 0; i < 8; ++i) {
    const int krow = kt * 16 + i + half * 8;
    O[(size_t)(bb * n + krow) * HD + h * 128 + dcol] = acc[i];
  }
}

// ---------------- gather masked rows ----------------------------------------
__global__ void gather_kernel(const float* __restrict__ X,
                              const int* __restrict__ idx,
                              float* __restrict__ HM) {
  const int rr = blockIdx.x;  // 0..63
  const int bb = rr >> 5, i = rr & 31;
  const int src = idx[bb * 32 + i];
  HM[(size_t)rr * 128 + threadIdx.x] =
      X[(size_t)(bb * 256 + src) * 128 + threadIdx.x];
}

// ---------------------------------------------------------------------------
extern "C" void kernel_launch(void* const* d_in, const int* in_sizes, int n_in,
                              void* d_out, int out_size, void* d_ws,
                              size_t ws_size, hipStream_t stream) {
  (void)in_sizes; (void)n_in; (void)out_size; (void)ws_size;

  const float* seqs  = (const float*)d_in[0];
  const float* dams  = (const float*)d_in[1];
  const int*   midx  = (const int*)d_in[2];
  const int*   amask = (const int*)d_in[3];
  const float* fe_w1 = (const float*)d_in[4];
  const float* fe_b1 = (const float*)d_in[5];
  const float* fe_w2 = (const float*)d_in[6];
  const float* fe_b2 = (const float*)d_in[7];
  const float* ln_g  = (const float*)d_in[8];
  const float* ln_b  = (const float*)d_in[9];
  const float* wq    = (const float*)d_in[10];
  const float* wk    = (const float*)d_in[11];
  const float* wv    = (const float*)d_in[12];
  const float* wo    = (const float*)d_in[13];
  const float* rpw1  = (const float*)d_in[14];
  const float* rpb1  = (const float*)d_in[15];
  const float* rpw2  = (const float*)d_in[16];
  const float* rpb2  = (const float*)d_in[17];
  const float* ln1g  = (const float*)d_in[18];
  const float* ln1b  = (const float*)d_in[19];
  const float* fw1   = (const float*)d_in[20];
  const float* fb1   = (const float*)d_in[21];
  const float* fw2   = (const float*)d_in[22];
  const float* fb2   = (const float*)d_in[23];
  const float* ln2g  = (const float*)d_in[24];
  const float* ln2b  = (const float*)d_in[25];
  const float* linw  = (const float*)d_in[26];
  const float* linb  = (const float*)d_in[27];
  const float* dw1   = (const float*)d_in[28];
  const float* db1   = (const float*)d_in[29];
  const float* dw2   = (const float*)d_in[30];
  const float* db2   = (const float*)d_in[31];

  float* ws = (float*)d_ws;
  float* X   = ws; ws += 512 * 128;   // activations (b*n, d)
  float* T   = ws; ws += 512 * 128;   // pre-LN temp
  float* Qb  = ws; ws += 512 * 512;   // q projection (b*n, h*dk)
  float* Kb  = ws; ws += 512 * 512;
  float* Vb  = ws; ws += 512 * 512;
  float* Ob  = ws; ws += 512 * 512;   // attention output (concat heads)
  float* Hb  = ws; ws += 512 * 512;   // ffn hidden
  float* RW  = ws; ws += 256;         // folded rp weight (2,128)
  float* RC  = ws; ws += 128;         // folded rp bias (128)
  float* HM  = ws; ws += 64 * 128;
  float* HM2 = ws; ws += 64 * 128;
  float* HM3 = ws; ws += 64 * 128;

  const dim3 blk(256);
  auto gg = [](int M, int N) { return dim3((N + 127) / 128, M / 16); };

  // --- feature embed: x = LN(sqrt(128) * ((seqs@fe_w1+b1)@fe_w2+b2)) ---
  gemm_kernel<0><<<gg(512, 128), blk, 0, stream>>>(seqs, fe_w1, fe_b1, T, 512, 128, 13);
  gemm_kernel<0><<<gg(512, 128), blk, 0, stream>>>(T, fe_w2, fe_b2, X, 512, 128, 128);
  ln_kernel<<<512, 128, 0, stream>>>(X, nullptr, 11.313708498984761f, ln_g, ln_b, X);

  for (int l = 0; l < 2; ++l) {
    const float* wql = wq + (size_t)l * 128 * 512;
    const float* wkl = wk + (size_t)l * 128 * 512;
    const float* wvl = wv + (size_t)l * 128 * 512;
    const float* wol = wo + (size_t)l * 512 * 128;

    gemm_kernel<0><<<gg(512, 512), blk, 0, stream>>>(X, wql, nullptr, Qb, 512, 512, 128);
    gemm_kernel<0><<<gg(512, 512), blk, 0, stream>>>(X, wkl, nullptr, Kb, 512, 512, 128);
    gemm_kernel<0><<<gg(512, 512), blk, 0, stream>>>(X, wvl, nullptr, Vb, 512, 512, 128);

    rpfold_kernel<<<1, 128, 0, stream>>>(rpw1 + l * 256, rpb1 + l * 128,
                                         rpw2 + l * 16384, rpb2 + l * 128, RW, RC);

    attn_kernel<<<dim3(16, 4, 2), blk, 0, stream>>>(Qb, Kb, Vb, dams, amask,
                                                    RW, RC, Ob);

    gemm_kernel<0><<<gg(512, 128), blk, 0, stream>>>(Ob, wol, nullptr, T, 512, 128, 512);
    ln_kernel<<<512, 128, 0, stream>>>(T, X, 1.f, ln1g + l * 128, ln1b + l * 128, X);

    gemm_kernel<1><<<gg(512, 512), blk, 0, stream>>>(X, fw1 + (size_t)l * 128 * 512,
                                                     fb1 + l * 512, Hb, 512, 512, 128);
    gemm_kernel<0><<<gg(512, 128), blk, 0, stream>>>(Hb, fw2 + (size_t)l * 512 * 128,
                                                     fb2 + l * 128, T, 512, 128, 512);
    ln_kernel<<<512, 128, 0, stream>>>(T, X, 1.f, ln2g + l * 128, ln2b + l * 128, X);
  }

  // --- decode head ---
  gather_kernel<<<64, 128, 0, stream>>>(X, midx, HM);
  gemm_kernel<2><<<gg(64, 128), blk, 0, stream>>>(HM, linw, linb, HM2, 64, 128, 128);
  ln_kernel<<<64, 128, 0, stream>>>(HM2, nullptr, 1.f, ln_g, ln_b, HM);
  gemm_kernel<0><<<gg(64, 128), blk, 0, stream>>>(HM, dw1, db1, HM3, 64, 128, 128);
  gemm_kernel<0><<<gg(64, 1), blk, 0, stream>>>(HM3, dw2, db2, (float*)d_out, 64, 1, 128);
}